// GATLayer_27419071218019
// MI455X (gfx1250) — compile-verified
//
#include <hip/hip_runtime.h>
#include <math.h>

#define IN_DIM   128
#define OUT_DIM  64
#define NEG_SLOPE 0.01f

typedef __attribute__((ext_vector_type(2))) float v2f;
typedef __attribute__((ext_vector_type(8))) float v8f;

// ---------- order-preserving float<->uint encoding for atomic max ----------
__device__ __forceinline__ unsigned enc_f32(float f) {
    unsigned u = __float_as_uint(f);
    return (u & 0x80000000u) ? ~u : (u | 0x80000000u);
}
__device__ __forceinline__ float dec_f32(unsigned u) {
    return __uint_as_float((u & 0x80000000u) ? (u & 0x7FFFFFFFu) : ~u);
}

// ---------- init: zero h_out, e_sum, e_max_enc ----------
__global__ void gat_init_kernel(float* __restrict__ h_out, int n_out,
                                unsigned* __restrict__ e_max_enc,
                                float* __restrict__ e_sum, int n_nodes) {
    int i = blockIdx.x * blockDim.x + threadIdx.x;
    int stride = gridDim.x * blockDim.x;
    for (int j = i; j < n_out; j += stride) h_out[j] = 0.0f;
    for (int j = i; j < n_nodes; j += stride) {
        e_max_enc[j] = 0u;      // encodes "below -inf"
        e_sum[j]     = 0.0f;
    }
}

// ---------- kernel 1: z = h @ fc_w^T via V_WMMA_F32_16X16X4_F32 ----------
// Block = 128 threads = 4 waves. Block computes 16 rows x 64 cols of z;
// wave w computes the 16x16 tile at columns [16w, 16w+16).
#define LDS_PITCH 132   // 128 + 4 pad -> bank-conflict-free column reads
__global__ __launch_bounds__(128) void gat_gemm_wmma_kernel(
        const float* __restrict__ h, const float* __restrict__ fc_w,
        float* __restrict__ z, int n_nodes) {
    __shared__ float lds_w[OUT_DIM * LDS_PITCH];

    const int tid = threadIdx.x;
    // stage fc_w (64x128 fp32, 32KB) into padded LDS, float4 granularity
    for (int i = tid; i < OUT_DIM * (IN_DIM / 4); i += 128) {
        int row = i >> 5;            // /32 float4 per row
        int seg = (i & 31) << 2;     // *4 floats
        const float4 v = *(const float4*)&fc_w[row * IN_DIM + seg];
        *(float4*)&lds_w[row * LDS_PITCH + seg] = v;
    }
    __syncthreads();

    const int wave   = tid >> 5;
    const int lane   = tid & 31;
    const int half   = lane >> 4;    // 0: K even pair base, 1: K odd pair base
    const int lid    = lane & 15;
    const int m_base = blockIdx.x * 16;
    const int n_base = wave * 16;

    int m_row = m_base + lid;
    if (m_row >= n_nodes) m_row = n_nodes - 1;   // safe clamp for loads

    const float* arow = h + (size_t)m_row * IN_DIM + 2 * half;
    const float* brow = lds_w + (n_base + lid) * LDS_PITCH + 2 * half;

    v8f c = {};
#pragma unroll
    for (int k0 = 0; k0 < IN_DIM; k0 += 4) {
        v2f a = *(const v2f*)(arow + k0);   // A[m, k0+2h : k0+2h+2]
        v2f b = *(const v2f*)(brow + k0);   // B[k0+2h : .., n] = fc_w[n, k..]
        c = __builtin_amdgcn_wmma_f32_16x16x4_f32(
                /*neg_a=*/false, a, /*neg_b=*/false, b,
                /*c_mod=*/(short)0, c, /*reuse_a=*/false, /*reuse_b=*/false);
    }

    // C/D layout: VGPR r -> row m_base + 8*half + r, col n_base + lid
#pragma unroll
    for (int r = 0; r < 8; ++r) {
        int m = m_base + half * 8 + r;
        if (m < n_nodes)
            z[(size_t)m * OUT_DIM + n_base + lid] = c[r];
    }
}

// ---------- kernel 2: per-node scores s_src = z.a_src, s_dst = z.a_dst ----------
// One wave per node; lane l covers columns 2l, 2l+1.
__global__ __launch_bounds__(256) void gat_scores_kernel(
        const float* __restrict__ z, const float* __restrict__ attn_w,
        float* __restrict__ s_src, float* __restrict__ s_dst, int n_nodes) {
    const int lane = threadIdx.x & 31;
    const int node = (blockIdx.x * blockDim.x + threadIdx.x) >> 5;
    if (node >= n_nodes) return;

    const v2f zv = *(const v2f*)&z[(size_t)node * OUT_DIM + 2 * lane];
    const v2f as = *(const v2f*)&attn_w[2 * lane];
    const v2f ad = *(const v2f*)&attn_w[OUT_DIM + 2 * lane];

    float ps = zv.x * as.x + zv.y * as.y;
    float pd = zv.x * ad.x + zv.y * ad.y;
#pragma unroll
    for (int m = 16; m >= 1; m >>= 1) {
        ps += __shfl_xor(ps, m, 32);
        pd += __shfl_xor(pd, m, 32);
    }
    if (lane == 0) {
        s_src[node] = ps;
        s_dst[node] = pd;
    }
}

// ---------- kernel 3: e = leaky_relu(s_src[src]+s_dst[dst]); segment max ----------
__global__ __launch_bounds__(256) void gat_edge_max_kernel(
        const int* __restrict__ src, const int* __restrict__ dst,
        const float* __restrict__ s_src, const float* __restrict__ s_dst,
        float* __restrict__ e, unsigned* __restrict__ e_max_enc, int n_edges) {
    int i = blockIdx.x * blockDim.x + threadIdx.x;
    if (i >= n_edges) return;
    float v = s_src[src[i]] + s_dst[dst[i]];
    v = (v > 0.0f) ? v : NEG_SLOPE * v;
    e[i] = v;
    atomicMax(&e_max_enc[dst[i]], enc_f32(v));
}

// ---------- kernel 4: e_sum[dst] += exp(e - e_max[dst]) ----------
__global__ __launch_bounds__(256) void gat_edge_sum_kernel(
        const int* __restrict__ dst, const float* __restrict__ e,
        const unsigned* __restrict__ e_max_enc, float* __restrict__ e_sum,
        int n_edges) {
    int i = blockIdx.x * blockDim.x + threadIdx.x;
    if (i >= n_edges) return;
    int d = dst[i];
    atomicAdd(&e_sum[d], __expf(e[i] - dec_f32(e_max_enc[d])));
}

// ---------- kernel 5: h_out[dst] += alpha * z[src]; one wave per edge ----------
__global__ __launch_bounds__(256) void gat_scatter_kernel(
        const int* __restrict__ src, const int* __restrict__ dst,
        const float* __restrict__ e, const unsigned* __restrict__ e_max_enc,
        const float* __restrict__ e_sum, const float* __restrict__ z,
        float* __restrict__ h_out, int n_edges) {
    const int lane = threadIdx.x & 31;
    const int edge = (blockIdx.x * blockDim.x + threadIdx.x) >> 5;
    if (edge >= n_edges) return;

    const int s = src[edge];
    const int d = dst[edge];
    const float alpha = __expf(e[edge] - dec_f32(e_max_enc[d])) / e_sum[d];

    const v2f zv = *(const v2f*)&z[(size_t)s * OUT_DIM + 2 * lane];
    float* out = &h_out[(size_t)d * OUT_DIM + 2 * lane];
    atomicAdd(&out[0], alpha * zv.x);
    atomicAdd(&out[1], alpha * zv.y);
}

extern "C" void kernel_launch(void* const* d_in, const int* in_sizes, int n_in,
                              void* d_out, int out_size, void* d_ws, size_t ws_size,
                              hipStream_t stream) {
    const float* h      = (const float*)d_in[0];
    const int*   src    = (const int*)d_in[1];
    const int*   dst    = (const int*)d_in[2];
    const float* fc_w   = (const float*)d_in[3];
    const float* attn_w = (const float*)d_in[4];
    float*       h_out  = (float*)d_out;

    const int n_nodes = in_sizes[0] / IN_DIM;
    const int n_edges = in_sizes[1];

    // workspace layout (16B aligned slabs)
    char* ws = (char*)d_ws;
    size_t off = 0;
    auto alloc = [&](size_t bytes) {
        void* p = ws + off;
        off += (bytes + 15) & ~size_t(15);
        return p;
    };
    float*    z         = (float*)alloc((size_t)n_nodes * OUT_DIM * sizeof(float));
    float*    s_src     = (float*)alloc((size_t)n_nodes * sizeof(float));
    float*    s_dst     = (float*)alloc((size_t)n_nodes * sizeof(float));
    unsigned* e_max_enc = (unsigned*)alloc((size_t)n_nodes * sizeof(unsigned));
    float*    e_sum     = (float*)alloc((size_t)n_nodes * sizeof(float));
    float*    e         = (float*)alloc((size_t)n_edges * sizeof(float));

    // init accumulators + output
    {
        int total = n_nodes * OUT_DIM;
        int blocks = (total + 255) / 256;
        if (blocks > 4096) blocks = 4096;
        gat_init_kernel<<<blocks, 256, 0, stream>>>(h_out, total, e_max_enc,
                                                    e_sum, n_nodes);
    }
    // GEMM: z = h @ fc_w^T
    {
        int blocks = (n_nodes + 15) / 16;
        gat_gemm_wmma_kernel<<<blocks, 128, 0, stream>>>(h, fc_w, z, n_nodes);
    }
    // per-node attention scalars
    {
        int blocks = (n_nodes * 32 + 255) / 256;
        gat_scores_kernel<<<blocks, 256, 0, stream>>>(z, attn_w, s_src, s_dst,
                                                      n_nodes);
    }
    // edge score + segment max
    {
        int blocks = (n_edges + 255) / 256;
        gat_edge_max_kernel<<<blocks, 256, 0, stream>>>(src, dst, s_src, s_dst,
                                                        e, e_max_enc, n_edges);
    }
    // segment sum of exp
    {
        int blocks = (n_edges + 255) / 256;
        gat_edge_sum_kernel<<<blocks, 256, 0, stream>>>(dst, e, e_max_enc,
                                                        e_sum, n_edges);
    }
    // weighted scatter-add
    {
        int blocks = (n_edges * 32 + 255) / 256;
        gat_scatter_kernel<<<blocks, 256, 0, stream>>>(src, dst, e, e_max_enc,
                                                       e_sum, z, h_out, n_edges);
    }
}